// MultiScaleSensor_644245094703
// MI455X (gfx1250) — compile-verified
//
#include <hip/hip_runtime.h>
#include <hip/hip_bf16.h>
#include <math.h>

// ---------------------------------------------------------------------------
// Multi-scale glimpse sensor for MI455X (gfx1250, wave32).
//
// out[b, 0, c, :, :] = 64x64 crop of img0 (zero pad)
// out[b, 1, c, :, :] = W2 * P128 * W2^T   (antialiased bilinear /2)
// out[b, 2, c, :, :] = W4 * P256 * W4^T   (antialiased bilinear /4)
//
// Separable antialiased resize expressed as two fp32 GEMMs executed with
// V_WMMA_F32_16X16X4_F32 (full fp32, RNE — matches reference numerics).
// Memory-bound (~66 MB gathered + 9.4 MB out => ~3-4 us at 23.3 TB/s);
// every cropped pixel is read from HBM exactly once, coalesced.
// OOB handling is branchless (clamped address + v_cndmask); the inner loop
// is a single uniform load -> 4x WMMA body so the accumulators stay pinned
// to one VGPR tuple (no allocator copy churn) and the compiler's scheduler
// pipelines the global loads across iterations itself.
// ---------------------------------------------------------------------------

typedef __attribute__((ext_vector_type(2))) float v2f;
typedef __attribute__((ext_vector_type(8))) float v8f;

#define IMG 512
#define C_CH 3
#define G 64

__device__ __forceinline__ int clampi(int v, int lo, int hi) {
    return v < lo ? lo : (v > hi ? hi : v);
}

// ---------------------------------------------------------------------------
// Build jax.image.resize(bilinear, antialias=True) weight matrices.
// W[j,i] = tri((i - x_j)/k) / rowsum,  x_j = (j+0.5)*k - 0.5,  k = 2 or 4.
// grid (64, 2), block 256.
// ---------------------------------------------------------------------------
__global__ void WeightsKernel(float* __restrict__ w2, float* __restrict__ w4) {
    const int j = blockIdx.x;            // output row 0..63
    const int s = blockIdx.y;            // 0 -> S=128 (k=2), 1 -> S=256 (k=4)
    const int S = s ? 256 : 128;
    const float k = s ? 4.0f : 2.0f;
    const float invk = 1.0f / k;
    const float x = (j + 0.5f) * k - 0.5f;
    float* __restrict__ w = s ? w4 : w2;

    // row sum (redundant per thread; trivial cost)
    float sum = 0.0f;
    for (int ii = 0; ii < S; ++ii) {
        float d = fabsf((float)ii - x) * invk;
        sum += fmaxf(0.0f, 1.0f - d);
    }
    int i = threadIdx.x;
    if (i < S) {
        float d = fabsf((float)i - x) * invk;
        float wt = fmaxf(0.0f, 1.0f - d);
        w[j * S + i] = wt / sum;
    }
}

__device__ __forceinline__ void decode_location(const float* __restrict__ loc,
                                                int b, int pad, int& gx0, int& gy0) {
    float lx = loc[b * 2 + 0];
    float ly = loc[b * 2 + 1];
    // trunc == floor for the non-negative range produced here (matches int32 cast)
    int stx = (int)(0.5f * ((lx + 1.0f) * (float)(IMG - 1)));
    int sty = (int)(0.5f * ((ly + 1.0f) * (float)(IMG - 1)));
    gx0 = stx - pad;
    gy0 = sty - pad;
}

// ---------------------------------------------------------------------------
// Slot 0: direct 64x64 crop, branchless OOB. grid = B*C, block = 256.
// ---------------------------------------------------------------------------
__global__ void CropKernel(const float* __restrict__ img0,
                           const float* __restrict__ loc,
                           float* __restrict__ out) {
    const int bc = blockIdx.x;
    const int b = bc / C_CH, c = bc % C_CH;
    int gx0, gy0;
    decode_location(loc, b, G / 2, gx0, gy0);

    const float* __restrict__ src = img0 + (size_t)(b * C_CH + c) * IMG * IMG;
    float* __restrict__ dst = out + ((size_t)(b * 3 + 0) * C_CH + c) * G * G;

    for (int t = threadIdx.x; t < G * G; t += blockDim.x) {
        int r = t >> 6, cc = t & 63;
        int gy = gy0 + r, gx = gx0 + cc;
        bool ok = ((unsigned)gy < IMG) && ((unsigned)gx < IMG);
        // unconditional load from clamped (always-valid) address, select zero
        float v = src[clampi(gy, 0, IMG - 1) * IMG + clampi(gx, 0, IMG - 1)];
        dst[t] = ok ? v : 0.0f;
    }
}

// ---------------------------------------------------------------------------
// Resize via two fp32 WMMA GEMMs.   S = patch size (128 or 256).
// Block = (S/16) waves. One block per (b,c).
//   GEMM1: T(64 x S) = W(64 x S) * P(S x S)   (P gathered from global)
//   GEMM2: out(64x64) = T * W^T               (from LDS)
//
// Fragment layouts per CDNA5 ISA 7.12.2 (32-bit, wave32):
//   A 16x4:  lanes 0-15 -> M=lane, K = {0,1}; lanes 16-31 -> M=lane-16, K={2,3}
//   B 4x16:  lanes 0-15 -> N=lane, K = {0,1}; lanes 16-31 -> N=lane-16, K={2,3}
//   C/D:     VGPR r: lanes 0-15 -> (M=r, N=lane); lanes 16-31 -> (M=r+8, N=lane-16)
// ---------------------------------------------------------------------------
template <int S>
__global__ void ResizeWmmaKernel(const float* __restrict__ img,
                                 const float* __restrict__ loc,
                                 const float* __restrict__ Wg,
                                 float* __restrict__ out,
                                 int slot) {
    constexpr int NW = S / 16;                 // waves per block
    extern __shared__ float smem[];
    float* __restrict__ wlds = smem;           // 64 * S  (weight matrix)
    float* __restrict__ tlds = smem + 64 * S;  // 64 * S  (intermediate T)

    const int bc = blockIdx.x;
    const int b = bc / C_CH, c = bc % C_CH;
    const int wave = threadIdx.x >> 5;
    const int lane = threadIdx.x & 31;
    const int half = lane >> 4;                // 0: K {0,1}, 1: K {2,3}
    const int ln = lane & 15;

    int gx0, gy0;
    decode_location(loc, b, S / 2, gx0, gy0);

    // stage resize weights into LDS (reused by both GEMMs, A and B operands)
    for (int idx = threadIdx.x; idx < 64 * S; idx += blockDim.x)
        wlds[idx] = Wg[idx];
    __syncthreads();

    const float* __restrict__ src = img + (size_t)(b * C_CH + c) * IMG * IMG;

    // ---------------- GEMM1: T = W * P -----------------------------------
    {
        const int n0 = wave * 16;              // this wave's 16 patch columns
        const int jcol = gx0 + n0 + ln;
        const bool jok = ((unsigned)jcol < IMG);
        const int jx = clampi(jcol, 0, IMG - 1);

        v8f acc[4] = {};                       // 4 M-tiles (64 rows of T)
        for (int k = 0; k < S; k += 4) {
            // branchless B-fragment gather: clamped load + cndmask zero
            const int i0 = k + 2 * half;       // patch rows for this lane-half
            const int gyA = gy0 + i0;
            const int gyB = gyA + 1;
            float va = src[clampi(gyA, 0, IMG - 1) * IMG + jx];
            float vb = src[clampi(gyB, 0, IMG - 1) * IMG + jx];
            v2f bf;
            bf.x = (jok && (unsigned)gyA < IMG) ? va : 0.0f;
            bf.y = (jok && (unsigned)gyB < IMG) ? vb : 0.0f;
#pragma unroll
            for (int m = 0; m < 4; ++m) {
                v2f af;
                af.x = wlds[(m * 16 + ln) * S + i0];
                af.y = wlds[(m * 16 + ln) * S + i0 + 1];
                acc[m] = __builtin_amdgcn_wmma_f32_16x16x4_f32(
                    false, af, false, bf, (short)0, acc[m], false, false);
            }
        }
#pragma unroll
        for (int m = 0; m < 4; ++m)
#pragma unroll
            for (int r = 0; r < 8; ++r)
                tlds[(m * 16 + r + 8 * half) * S + n0 + ln] = acc[m][r];
    }
    __syncthreads();

    // ---------------- GEMM2: out = T * W^T -------------------------------
    float* __restrict__ dst = out + ((size_t)(b * 3 + slot) * C_CH + c) * G * G;
    for (int t = wave; t < 16; t += NW) {      // 4x4 output tiles
        const int m0 = (t >> 2) * 16;
        const int n0 = (t & 3) * 16;
        v8f acc = {};
        for (int k = 0; k < S; k += 4) {
            const int kk = k + 2 * half;
            v2f af, bf;
            af.x = tlds[(m0 + ln) * S + kk];
            af.y = tlds[(m0 + ln) * S + kk + 1];
            // B[k,n] = W[n,k]  (transposed weights)
            bf.x = wlds[(n0 + ln) * S + kk];
            bf.y = wlds[(n0 + ln) * S + kk + 1];
            acc = __builtin_amdgcn_wmma_f32_16x16x4_f32(
                false, af, false, bf, (short)0, acc, false, false);
        }
#pragma unroll
        for (int r = 0; r < 8; ++r)
            dst[(m0 + r + 8 * half) * G + n0 + ln] = acc[r];
    }
}

// ---------------------------------------------------------------------------
extern "C" void kernel_launch(void* const* d_in, const int* in_sizes, int n_in,
                              void* d_out, int out_size, void* d_ws, size_t ws_size,
                              hipStream_t stream) {
    (void)in_sizes; (void)n_in; (void)out_size; (void)ws_size;
    const float* img0 = (const float*)d_in[0];
    const float* img2 = (const float*)d_in[1];
    const float* img4 = (const float*)d_in[2];
    const float* loc  = (const float*)d_in[3];
    float* out = (float*)d_out;

    float* w2 = (float*)d_ws;            // 64*128 floats
    float* w4 = w2 + 64 * 128;           // 64*256 floats (96 KB total)

    const int B = 64;
    const int nBC = B * C_CH;            // 192 blocks

    // 1) build resize weight matrices (stream-ordered before consumers)
    WeightsKernel<<<dim3(64, 2), 256, 0, stream>>>(w2, w4);

    // 2) slot 0: raw crop
    CropKernel<<<nBC, 256, 0, stream>>>(img0, loc, out);

    // 3) slot 1: 128->64 resize, 8 waves/block, 64 KB LDS
    ResizeWmmaKernel<128><<<nBC, 8 * 32, 128 * 128 * sizeof(float), stream>>>(
        img2, loc, w2, out, 1);

    // 4) slot 2: 256->64 resize, 16 waves/block, 128 KB LDS
    ResizeWmmaKernel<256><<<nBC, 16 * 32, 128 * 256 * sizeof(float), stream>>>(
        img4, loc, w4, out, 2);
}